// GCNConvNet_27762668601833
// MI455X (gfx1250) — compile-verified
//
#include <hip/hip_runtime.h>

typedef __attribute__((ext_vector_type(2))) float v2f;
typedef __attribute__((ext_vector_type(8))) float v8f;

// ---------------------------------------------------------------------------
// Degree / normalization
// ---------------------------------------------------------------------------
__global__ void zero_cnt_kernel(int* __restrict__ cnt, int n) {
    int tid = blockIdx.x * blockDim.x + threadIdx.x;
    if (tid < n) cnt[tid] = 0;
}

__global__ void count_deg_kernel(const int* __restrict__ dst, int* __restrict__ cnt, int e) {
    int tid = blockIdx.x * blockDim.x + threadIdx.x;
    if (tid < e) atomicAdd(&cnt[dst[tid]], 1);
}

__global__ void calc_dis_kernel(const int* __restrict__ cnt, float* __restrict__ dis, int n) {
    int tid = blockIdx.x * blockDim.x + threadIdx.x;
    if (tid < n) {
        float deg = (float)(cnt[tid] + 1);   // +1 self loop; always >= 1
        dis[tid] = 1.0f / sqrtf(deg);
    }
}

// ---------------------------------------------------------------------------
// fp32 WMMA GEMM: C[M x NCOL] = (relu?)A[M x K] * B[K x NCOL]
// One wave32 computes MT row-tiles (16 rows each) x one 16-col tile, so each
// B fragment (2 loads) feeds MT v_wmma_f32_16x16x4_f32 ops. K/NCOL are
// template constants and the K-loop fully unrolls, so every global load uses
// an immediate ioffset (A subtile stride 16*K*4, B row stride NCOL*4).
//
// Fragment layouts (ISA 7.12.2):
//  A 16x4 : lane l -> row l&15, K = kbase + (l>>4)*2 .. +1  (2 VGPRs, v2f)
//  B 4x16 : lane l -> col l&15, K = kbase + (l>>4)*2 .. +1  (2 VGPRs)
//  C 16x16: VGPR i -> M = i + 8*(l>=16), N = l&15
// ---------------------------------------------------------------------------
template <int K, int NCOL, bool RELU, int MT>
__global__ __launch_bounds__(256) void gemm_wmma_kernel(
    const float* __restrict__ A, const float* __restrict__ B,
    float* __restrict__ C, int M) {
    constexpr int TILES_N = NCOL / 16;
    int wave = (blockIdx.x * blockDim.x + threadIdx.x) >> 5;
    int lane = threadIdx.x & 31;
    int tg = wave / TILES_N;             // row-tile group (MT tiles of 16 rows)
    int tn = wave - tg * TILES_N;        // col tile
    int rm = tg * (16 * MT);             // first row of this wave's group
    if (rm >= M) return;                 // wave-uniform exit

    int half = lane >> 4;                // 0: K+0/1, 1: K+2/3
    int l15  = lane & 15;
    int col  = tn * 16 + l15;

    // Base pointers with lane-dependent parts folded in once.
    const float* abase = A + (size_t)(rm + l15) * K + half * 2;
    const float* bbase = B + col + half * 2 * NCOL;

    v8f acc[MT];
    #pragma unroll
    for (int t = 0; t < MT; ++t) acc[t] = (v8f){};

    #pragma unroll
    for (int k = 0; k < K; k += 4) {
        v2f bv;
        bv.x = bbase[k * NCOL];          // immediate offsets after unroll
        bv.y = bbase[k * NCOL + NCOL];
        #pragma unroll
        for (int t = 0; t < MT; ++t) {
            v2f av = *(const v2f*)(abase + t * 16 * K + k);
            if (RELU) { av.x = fmaxf(av.x, 0.0f); av.y = fmaxf(av.y, 0.0f); }
            acc[t] = __builtin_amdgcn_wmma_f32_16x16x4_f32(
                false, av, false, bv, (short)0, acc[t], false, false);
        }
    }

    #pragma unroll
    for (int t = 0; t < MT; ++t) {
        float* crow = C + (size_t)(rm + t * 16 + half * 8) * NCOL + col;
        #pragma unroll
        for (int i = 0; i < 8; ++i) crow[i * NCOL] = acc[t][i];
    }
}

// ---------------------------------------------------------------------------
// agg[i][j] = bias[j] + h[i][j] * dis[i]^2   (self-loop term + bias)
// ---------------------------------------------------------------------------
__global__ void init_agg_kernel(const float* __restrict__ h, const float* __restrict__ bias,
                                const float* __restrict__ dis, float* __restrict__ agg,
                                int total, int dshift) {
    int tid = blockIdx.x * blockDim.x + threadIdx.x;
    if (tid >= total) return;
    int i = tid >> dshift;
    int j = tid & ((1 << dshift) - 1);
    float d = dis[i];
    agg[tid] = bias[j] + h[tid] * d * d;
}

// ---------------------------------------------------------------------------
// Per-edge scatter: agg[dst] += h[src] * dis[src]*dis[dst]
// float4 gather from L2-resident h + 4x global_atomic_add_f32.
// ---------------------------------------------------------------------------
__global__ void scatter_edges_kernel(const float* __restrict__ h,
                                     const int* __restrict__ src,
                                     const int* __restrict__ dst,
                                     const float* __restrict__ dis,
                                     float* __restrict__ agg,
                                     int nedges, int cshift, int dshift) {
    int tid = blockIdx.x * blockDim.x + threadIdx.x;
    int total = nedges << cshift;
    if (tid >= total) return;
    int e = tid >> cshift;
    int c = (tid & ((1 << cshift) - 1)) << 2;
    int s = src[e];
    int d = dst[e];
    float nrm = dis[s] * dis[d];
    const float4* hp = (const float4*)(h + ((size_t)s << dshift) + c);
    float4 v = *hp;
    float* ap = agg + ((size_t)d << dshift) + c;
    atomicAdd(ap + 0, v.x * nrm);
    atomicAdd(ap + 1, v.y * nrm);
    atomicAdd(ap + 2, v.z * nrm);
    atomicAdd(ap + 3, v.w * nrm);
}

// ---------------------------------------------------------------------------
// Host launcher
// ---------------------------------------------------------------------------
static inline int ceil_div(long long a, int b) { return (int)((a + b - 1) / b); }

extern "C" void kernel_launch(void* const* d_in, const int* in_sizes, int n_in,
                              void* d_out, int out_size, void* d_ws, size_t ws_size,
                              hipStream_t stream) {
    const float* x  = (const float*)d_in[0];
    const int*   ei = (const int*)d_in[1];
    const float* W0 = (const float*)d_in[2];
    const float* b0 = (const float*)d_in[3];
    const float* W1 = (const float*)d_in[4];
    const float* b1 = (const float*)d_in[5];
    const float* W2 = (const float*)d_in[6];
    const float* b2 = (const float*)d_in[7];
    float* out = (float*)d_out;

    const int IN_DIM = 128, HID = 128, OUT_D = 64;
    const int MT = 5;                 // row tiles per wave; 50000 % (16*5) == 0
    int Nn = in_sizes[0] / IN_DIM;    // 50000
    int E  = in_sizes[1] / 2;         // 800000
    const int* src = ei;
    const int* dst = ei + E;

    float* ws   = (float*)d_ws;
    float* dis  = ws;                          // N floats
    int*   cnt  = (int*)(ws + Nn);             // N ints
    float* bufA = ws + 2 * (size_t)Nn;         // N*128 floats
    float* bufB = bufA + (size_t)Nn * HID;     // N*128 floats

    const int T = 256;

    // degree -> dis
    zero_cnt_kernel<<<ceil_div(Nn, T), T, 0, stream>>>(cnt, Nn);
    count_deg_kernel<<<ceil_div(E, T), T, 0, stream>>>(dst, cnt, E);
    calc_dis_kernel<<<ceil_div(Nn, T), T, 0, stream>>>(cnt, dis, Nn);

    int groups = ceil_div(Nn, 16 * MT);                 // 625
    int blocks128 = ceil_div((long long)groups * (HID / 16) * 32, T);
    int blocks64  = ceil_div((long long)groups * (OUT_D / 16) * 32, T);

    // ---- Layer 0: h = x @ W0 ; agg = b0 + h*dis^2 + scatter ----
    gemm_wmma_kernel<128, 128, false, MT><<<blocks128, T, 0, stream>>>(x, W0, bufA, Nn);
    init_agg_kernel<<<ceil_div((long long)Nn * HID, T), T, 0, stream>>>(bufA, b0, dis, bufB, Nn * HID, 7);
    scatter_edges_kernel<<<ceil_div((long long)E * (HID / 4), T), T, 0, stream>>>(
        bufA, src, dst, dis, bufB, E, 5, 7);

    // ---- Layer 1: h = relu(agg) @ W1 ----
    gemm_wmma_kernel<128, 128, true, MT><<<blocks128, T, 0, stream>>>(bufB, W1, bufA, Nn);
    init_agg_kernel<<<ceil_div((long long)Nn * HID, T), T, 0, stream>>>(bufA, b1, dis, bufB, Nn * HID, 7);
    scatter_edges_kernel<<<ceil_div((long long)E * (HID / 4), T), T, 0, stream>>>(
        bufA, src, dst, dis, bufB, E, 5, 7);

    // ---- Layer 2: h = relu(agg) @ W2 ; output directly ----
    gemm_wmma_kernel<128, 64, true, MT><<<blocks64, T, 0, stream>>>(bufB, W2, bufA, Nn);
    init_agg_kernel<<<ceil_div((long long)Nn * OUT_D, T), T, 0, stream>>>(bufA, b2, dis, out, Nn * OUT_D, 6);
    scatter_edges_kernel<<<ceil_div((long long)E * (OUT_D / 4), T), T, 0, stream>>>(
        bufA, src, dst, dis, out, E, 4, 6);
}